// CircuitModel_86818468921777
// MI455X (gfx1250) — compile-verified
//
#include <hip/hip_runtime.h>
#include <stdint.h>

// Problem constants (from reference)
#define N_INPUTS 256
#define N_REGS   256
#define BATCH_SZ 4096
#define T_STEPS  64

#define NB       16      // batch tile per workgroup
#define THREADS  256     // 8 waves (wave32)

typedef __attribute__((address_space(1))) int g_int;  // global
typedef __attribute__((address_space(3))) int l_int;  // LDS

// ---- CDNA5 async global->LDS copy (ASYNCcnt path) ----------------------
__device__ __forceinline__ void async_cp_f32(const float* g, float* l) {
#if __has_builtin(__builtin_amdgcn_global_load_async_to_lds_b32)
  __builtin_amdgcn_global_load_async_to_lds_b32(
      (g_int*)g, (l_int*)l, /*imm offset=*/0, /*cpol=*/0);
#else
  asm volatile("global_load_async_to_lds_b32 %0, %1, off"
               :
               : "v"((uint32_t)(uintptr_t)l), "v"(g)
               : "memory");
#endif
}

__device__ __forceinline__ void wait_async0() {
#if __has_builtin(__builtin_amdgcn_s_wait_asynccnt)
  __builtin_amdgcn_s_wait_asynccnt(0);
#else
  asm volatile("s_wait_asynccnt 0" ::: "memory");
#endif
}

// LDS layout (floats): X0[0,4096) S0[4096,8192) X1[8192,12288) S1[12288,16384)
// Segment pair (X,S) of the same parity is contiguous, so a gather index
// idx in [0,512) resolves to   parity*8192 + idx*16 + b   with no branch.
__global__ __launch_bounds__(THREADS) void circuit_softxor_kernel(
    const float* __restrict__ P,   // (256, 4096, 64) f32
    const int*   __restrict__ wa,  // (256,)
    const int*   __restrict__ wb,  // (256,)
    float*       __restrict__ out) // (4096, 256) f32
{
  __shared__ float lds[16384];

  const int tid = threadIdx.x;
  const int b   = tid & (NB - 1);   // batch lane within tile
  const int rg  = tid >> 4;         // register group (0..15)
  const int b0  = blockIdx.x * NB;

  // Per-thread gather offsets (float units within an 8192-float segment pair)
  // and global float-offsets of the 16 elements this thread streams per step.
  int offA[16], offB[16];
  uint32_t goff[16];
#pragma unroll
  for (int k = 0; k < 16; ++k) {
    const int r = rg + 16 * k;            // register handled at slot k
    offA[k] = wa[r] * NB + b;
    offB[k] = wb[r] * NB + b;
    const int e  = tid + THREADS * k;     // element id: e = i*16 + bb
    const int i  = e >> 4;
    const int bb = e & (NB - 1);
    goff[k] = (uint32_t)i * (BATCH_SZ * T_STEPS) + (uint32_t)(b0 + bb) * T_STEPS;
  }

  // init S0 = 0.5
#pragma unroll
  for (int k = 0; k < 16; ++k)
    lds[4096 + tid + THREADS * k] = 0.5f;

  // async prefetch of t=0 slab into X0
#pragma unroll
  for (int k = 0; k < 16; ++k)
    async_cp_f32(P + goff[k], &lds[tid + THREADS * k]);

  auto step = [&](int t, int cur, int nxt) {
    const int rbase  = cur * 8192;           // X[cur] + S[cur] (contiguous)
    const int xnbase = nxt * 8192;           // X[nxt] async target
    const int wbase  = nxt * 8192 + 4096;    // S[nxt] write region

    wait_async0();        // my X[cur] async loads landed
    __syncthreads();      // everyone's landed; also makes S[cur] stores visible

    // kick off async loads of the next timestep's slab as early as possible:
    // overlaps the DMA with both the sigmoid pass and the gather pass.
    // (All readers of X[nxt] finished before the barrier above.)
    if (t + 1 < T_STEPS) {
#pragma unroll
      for (int k = 0; k < 16; ++k)
        async_cp_f32(P + goff[k] + (t + 1), &lds[xnbase + tid + THREADS * k]);
    }

    // sigmoid(4*clip(x,-2,2)) in place on X[cur]
#pragma unroll
    for (int k = 0; k < 16; ++k) {
      const int e = tid + THREADS * k;
      float v = lds[rbase + e];
      v = fminf(fmaxf(v, -2.0f), 2.0f);
      lds[rbase + e] = __builtin_amdgcn_rcpf(1.0f + __expf(-4.0f * v));
    }
    __syncthreads();

    // gather + soft-XOR update:  s' = a + b - 2ab = fma(b, fma(-2,a,1), a)
    // S[nxt] writes become visible via the two barriers of the next step.
#pragma unroll
    for (int k = 0; k < 16; ++k) {
      const float av = lds[rbase + offA[k]];
      const float bv = lds[rbase + offB[k]];
      const float nv = fmaf(bv, fmaf(-2.0f, av, 1.0f), av);
      lds[wbase + rg * NB + b + 256 * k] = nv;   // S[nxt][r][b], r = rg+16k
    }
  };

  for (int t = 0; t < T_STEPS; t += 2) {
    step(t,     0, 1);
    step(t + 1, 1, 0);
  }

  __syncthreads();  // make final S0 stores visible to all threads

  // t=63 wrote S0 (segment [4096,8192)), layout [r][b] -> out[b][r]
#pragma unroll
  for (int bb = 0; bb < NB; ++bb)
    out[(uint64_t)(b0 + bb) * N_REGS + tid] = lds[4096 + tid * NB + bb];
}

extern "C" void kernel_launch(void* const* d_in, const int* in_sizes, int n_in,
                              void* d_out, int out_size, void* d_ws, size_t ws_size,
                              hipStream_t stream) {
  (void)in_sizes; (void)n_in; (void)out_size; (void)d_ws; (void)ws_size;
  const float* P  = (const float*)d_in[0];
  const int*   wa = (const int*)d_in[1];
  const int*   wb = (const int*)d_in[2];
  float*       out = (float*)d_out;

  circuit_softxor_kernel<<<BATCH_SZ / NB, THREADS, 0, stream>>>(P, wa, wb, out);
}